// CVMerge_41472204210311
// MI455X (gfx1250) — compile-verified
//
#include <hip/hip_runtime.h>
#include <stdint.h>

// gfx1250 async-DMA builtins are typed as pointers to a 16-byte int vector.
typedef int v4i __attribute__((vector_size(4 * sizeof(int))));
typedef __attribute__((address_space(1))) v4i* as1_v4i; // global (== __device__)
typedef __attribute__((address_space(3))) v4i* as3_v4i; // LDS    (== __shared__)

static constexpr int TPB  = 256;        // 8 wave32 per block
static constexpr int RPT  = 8;          // fold entries (rows) per thread
static constexpr int TILE = TPB * RPT;  // 2048 rows per block
static constexpr int DF   = 32;         // floats per row (D), 128 B/row

// ---------------- CDNA5 async global<->LDS helpers (ASYNCcnt) ----------------

__device__ __forceinline__ void async_wait0() {
#if __has_builtin(__builtin_amdgcn_s_wait_asynccnt)
  __builtin_amdgcn_s_wait_asynccnt(0);
#else
  asm volatile("s_wait_asynccnt 0x0" ::: "memory");
#endif
}

__device__ __forceinline__ void async_load16(const void* gsrc, void* lds_dst) {
#if __has_builtin(__builtin_amdgcn_global_load_async_to_lds_b128)
  __builtin_amdgcn_global_load_async_to_lds_b128(
      (as1_v4i)(uintptr_t)gsrc,
      (as3_v4i)(uint32_t)(uintptr_t)lds_dst, 0, 0);
#else
  uint32_t l = (uint32_t)(uintptr_t)lds_dst; // low 32 bits of generic = LDS addr
  uint64_t g = (uint64_t)(uintptr_t)gsrc;
  asm volatile("global_load_async_to_lds_b128 %0, %1, off"
               :: "v"(l), "v"(g) : "memory");
#endif
}

__device__ __forceinline__ void async_store16(void* gdst, const void* lds_src) {
#if __has_builtin(__builtin_amdgcn_global_store_async_from_lds_b128)
  __builtin_amdgcn_global_store_async_from_lds_b128(
      (as1_v4i)(uintptr_t)gdst,
      (as3_v4i)(uint32_t)(uintptr_t)lds_src, 0, 0);
#else
  uint32_t l = (uint32_t)(uintptr_t)lds_src;
  uint64_t g = (uint64_t)(uintptr_t)gdst;
  asm volatile("global_store_async_from_lds_b128 %0, %1, off"
               :: "v"(g), "v"(l) : "memory");
#endif
}

// ---------------- Kernel 1: per-block per-fold counts ----------------

extern "C" __global__ __launch_bounds__(TPB)
void cv_fold_count(const int* __restrict__ fold, int4* __restrict__ counts) {
  __shared__ int4 red[TPB];
  const int tid = threadIdx.x;
  const long base = (long)blockIdx.x * TILE + (long)tid * RPT;
  const int4* fp = reinterpret_cast<const int4*>(fold + base);
  const int4 f0 = fp[0], f1 = fp[1];
  const int fv[RPT] = {f0.x, f0.y, f0.z, f0.w, f1.x, f1.y, f1.z, f1.w};
  int4 c = make_int4(0, 0, 0, 0);
#pragma unroll
  for (int j = 0; j < RPT; ++j) {
    c.x += (fv[j] == 0); c.y += (fv[j] == 1);
    c.z += (fv[j] == 2); c.w += (fv[j] == 3);
  }
  red[tid] = c;
  __syncthreads();
#pragma unroll
  for (int s = TPB >> 1; s > 0; s >>= 1) {
    if (tid < s) {
      const int4 o = red[tid + s];
      c.x += o.x; c.y += o.y; c.z += o.z; c.w += o.w;
      red[tid] = c;
    }
    __syncthreads();
  }
  if (tid == 0) counts[blockIdx.x] = c;
}

// ---------------- Kernel 2: exclusive scan of block counts (1 block) --------

extern "C" __global__ __launch_bounds__(1024)
void cv_fold_scan(const int4* __restrict__ counts, int4* __restrict__ offsets,
                  int nblk) {
  __shared__ int4 sc[1024];
  const int tid = threadIdx.x;
  const int4 v = (tid < nblk) ? counts[tid] : make_int4(0, 0, 0, 0);
  int4 run = v;
  sc[tid] = run;
  __syncthreads();
#pragma unroll
  for (int off = 1; off < 1024; off <<= 1) {
    int4 t = make_int4(0, 0, 0, 0);
    if (tid >= off) t = sc[tid - off];
    __syncthreads();
    run.x += t.x; run.y += t.y; run.z += t.z; run.w += t.w;
    sc[tid] = run;
    __syncthreads();
  }
  if (tid < nblk)
    offsets[tid] = make_int4(run.x - v.x, run.y - v.y, run.z - v.z, run.w - v.w);
}

// ---------------- Kernel 3: rank + gather via async LDS staging -------------

extern "C" __global__ __launch_bounds__(TPB)
void cv_fold_gather(const float* __restrict__ x0, const float* __restrict__ x1,
                    const float* __restrict__ x2, const float* __restrict__ x3,
                    const int* __restrict__ fold,
                    const int4* __restrict__ offsets,
                    float* __restrict__ out) {
  __shared__ int4     scanb[TPB];    // 4 KB: intra-block scan
  __shared__ uint32_t rowsrc[TILE];  // 8 KB: packed (fold<<28 | srcRow)
  __shared__ float4   stage[TPB];    // 4 KB: async-DMA staging, 16 B/lane

  const int tid = threadIdx.x;
  const long gbase = (long)blockIdx.x * TILE;

  // Recompute this thread's per-fold counts over its 8 consecutive rows.
  const int4* fp = reinterpret_cast<const int4*>(fold + gbase + (long)tid * RPT);
  const int4 f0 = fp[0], f1 = fp[1];
  const int fv[RPT] = {f0.x, f0.y, f0.z, f0.w, f1.x, f1.y, f1.z, f1.w};
  int4 c = make_int4(0, 0, 0, 0);
#pragma unroll
  for (int j = 0; j < RPT; ++j) {
    c.x += (fv[j] == 0); c.y += (fv[j] == 1);
    c.z += (fv[j] == 2); c.w += (fv[j] == 3);
  }

  // Intra-block inclusive scan (int4 = all 4 folds at once).
  int4 run = c;
  scanb[tid] = run;
  __syncthreads();
#pragma unroll
  for (int off = 1; off < TPB; off <<= 1) {
    int4 t = make_int4(0, 0, 0, 0);
    if (tid >= off) t = scanb[tid - off];
    __syncthreads();
    run.x += t.x; run.y += t.y; run.z += t.z; run.w += t.w;
    scanb[tid] = run;
    __syncthreads();
  }

  // Thread's starting rank per fold = global block offset + exclusive prefix.
  const int4 bo = offsets[blockIdx.x];
  int nxt[4] = { bo.x + run.x - c.x, bo.y + run.y - c.y,
                 bo.z + run.z - c.z, bo.w + run.w - c.w };
#pragma unroll
  for (int j = 0; j < RPT; ++j) {
    const int f = fv[j] & 3;
    const uint32_t r = (uint32_t)nxt[f]++;
    rowsrc[tid * RPT + j] = ((uint32_t)f << 28) | r;
  }
  __syncthreads();

  // Cooperative copy: 8 lanes per 128-B row, 4 rows per wave per pass.
  // Each lane moves 16 B via async global->LDS->global DMA (ASYNCcnt).
  const float* xs[4] = {x0, x1, x2, x3};
  const int grp = tid >> 3;   // row slot within the 32 rows of a pass
  const int sub = tid & 7;    // 16-B chunk within the row
#pragma unroll 1
  for (int p = 0; p < TILE / 32; ++p) {
    const int row = p * 32 + grp;
    const uint32_t pr = rowsrc[row];
    const int  f = (int)(pr >> 28);
    const long r = (long)(pr & 0x0FFFFFFFu);
    const float* src = xs[f] + (r << 5) + (sub << 2);           // x_f row chunk
    float*       dst = out + ((gbase + row) << 5) + (sub << 2); // out row chunk
    async_load16(src, &stage[tid]);
    async_wait0();                       // data landed in LDS
    async_store16(dst, &stage[tid]);
    async_wait0();                       // stage slot free for next pass
  }
}

// ---------------- host-side launch ----------------

extern "C" void kernel_launch(void* const* d_in, const int* in_sizes, int n_in,
                              void* d_out, int out_size, void* d_ws, size_t ws_size,
                              hipStream_t stream) {
  (void)n_in; (void)out_size; (void)ws_size;
  const float* x0  = (const float*)d_in[0];
  const float* x1  = (const float*)d_in[1];
  const float* x2  = (const float*)d_in[2];
  const float* x3  = (const float*)d_in[3];
  const int*   fold = (const int*)d_in[4];
  float* out = (float*)d_out;

  const int n    = in_sizes[4];     // total rows N (2,097,152)
  const int nblk = n / TILE;        // 1024 blocks of 2048 rows

  int4* counts  = (int4*)d_ws;      // nblk * 16 B
  int4* offsets = counts + nblk;    // nblk * 16 B  (32 KB total scratch)

  cv_fold_count <<<nblk, TPB,  0, stream>>>(fold, counts);
  cv_fold_scan  <<<1,    1024, 0, stream>>>(counts, offsets, nblk);
  cv_fold_gather<<<nblk, TPB,  0, stream>>>(x0, x1, x2, x3, fold, offsets, out);
}